// MultiheadAttention_44976897523740
// MI455X (gfx1250) — compile-verified
//
#include <hip/hip_runtime.h>
#include <hip/hip_bf16.h>

typedef __attribute__((ext_vector_type(2))) float v2f;
typedef __attribute__((ext_vector_type(8))) float v8f;
typedef __attribute__((ext_vector_type(4))) int v4i;

#define NB 16
#define QL 77
#define KL 1576
#define NH 12
#define HD 64
#define NF 8
#define FJ 197           // KL / NUM_FRAMES
#define EMB 768
#define PROJ_N 1536
#define NKT 99           // ceil(KL/16)
#define QTILES 5         // ceil(QL/16)
#define NWAVES 4
#define ITERS 25         // ceil(NKT/NWAVES)

#if defined(__gfx1250__) && __has_builtin(__builtin_amdgcn_global_load_async_to_lds_b128)
#define USE_ASYNC 1
#else
#define USE_ASYNC 0
#endif

// async-copy payload pointers: 128-bit = 4 x i32, global (AS1) src, LDS (AS3) dst
typedef __attribute__((address_space(1))) v4i gv4i_t;
typedef __attribute__((address_space(3))) v4i lv4i_t;

__device__ __forceinline__ void wait_async0() {
#if __has_builtin(__builtin_amdgcn_s_wait_asynccnt)
  __builtin_amdgcn_s_wait_asynccnt(0);
#else
  asm volatile("s_wait_asynccnt 0x0" ::: "memory");
#endif
}

// ---- order-preserving float<->uint transform (for ds_max_u32 softmax maxes) ----
__device__ __forceinline__ unsigned f2ord(float f) {
  unsigned u = __float_as_uint(f);
  return (u & 0x80000000u) ? ~u : (u | 0x80000000u);
}
__device__ __forceinline__ float ord2f(unsigned o) {
  unsigned u = (o & 0x80000000u) ? (o ^ 0x80000000u) : ~o;
  return __uint_as_float(u);
}

// 16x16 logits tile via 16 chained V_WMMA_F32_16X16X4_F32 (K = 64 channels).
// A = qbuf rows (M=lane&15), B = k^T: both read as (lane&15)*64 + 4*ks + 2*(lane>>4).
__device__ __forceinline__ v8f qk_tile16(const float* __restrict__ qbuf,
                                         const float* __restrict__ kbuf,
                                         int lane) {
  const int row = (lane & 15) * HD;
  const int off = (lane >> 4) * 2;
  v8f c = {};
#pragma unroll
  for (int ks = 0; ks < 16; ++ks) {
    v2f a = *(const v2f*)(qbuf + row + ks * 4 + off);
    v2f b = *(const v2f*)(kbuf + row + ks * 4 + off);
    c = __builtin_amdgcn_wmma_f32_16x16x4_f32(false, a, false, b, (short)0, c,
                                              false, false);
  }
  return c;
}

// Stage 16 rows x 64 f32 (one k/v tile for one head) global -> LDS, zero-pad past KL.
// Uses GLOBAL_LOAD_ASYNC_TO_LDS_B128 (ASYNCcnt) when available: no VGPR round-trip.
__device__ __forceinline__ void stage_rows(const float* __restrict__ gbase,
                                           int kk0, float* __restrict__ dst,
                                           int lane) {
  const int r = lane & 15;
  const int c0 = (lane >> 4) * 32;
  float* d = dst + r * HD + c0;
  const int kk = kk0 + r;
  if (kk < KL) {
    const float* s = gbase + (size_t)kk * (NH * HD) + c0;
#if USE_ASYNC
#pragma unroll
    for (int u = 0; u < 8; ++u)
      __builtin_amdgcn_global_load_async_to_lds_b128(
          (gv4i_t*)(s + 4 * u), (lv4i_t*)(d + 4 * u), 0, 0);
#else
    float4* d4 = (float4*)d;
    const float4* s4 = (const float4*)s;
#pragma unroll
    for (int u = 0; u < 8; ++u) d4[u] = s4[u];
#endif
  } else {
    const float4 z = make_float4(0.f, 0.f, 0.f, 0.f);
    float4* d4 = (float4*)d;
#pragma unroll
    for (int u = 0; u < 8; ++u) d4[u] = z;
  }
#if USE_ASYNC
  wait_async0();
#endif
}

// ---------------- generic fp32 WMMA GEMM: C = A(MxK) @ B(KxN) + bias ----------------
__global__ __launch_bounds__(256) void gemm16_bias_kernel(
    const float* __restrict__ A, const float* __restrict__ B,
    const float* __restrict__ bias, float* __restrict__ C, int M, int N, int K) {
  const int lane = threadIdx.x & 31;
  const int wave = threadIdx.x >> 5;
  const int tile = blockIdx.x * 8 + wave;
  const int ntiles = N >> 4;
  if (tile >= (M >> 4) * ntiles) return;
  const int mt = tile / ntiles;
  const int nt = tile - mt * ntiles;
  const int half = lane >> 4;
  const int row = lane & 15;
  const int off = half * 2;
  const float* arow = A + (size_t)(mt * 16 + row) * K;
  const float* bcol = B + nt * 16 + row;
  v8f c = {};
#pragma unroll 4
  for (int ks = 0; ks < (K >> 2); ++ks) {
    const int ka = ks * 4 + off;
    v2f a = *(const v2f*)(arow + ka);
    v2f b;
    b.x = bcol[(size_t)ka * N];
    b.y = bcol[(size_t)(ka + 1) * N];
    c = __builtin_amdgcn_wmma_f32_16x16x4_f32(false, a, false, b, (short)0, c,
                                              false, false);
  }
  const float bs = bias[nt * 16 + row];
#pragma unroll
  for (int j = 0; j < 8; ++j) {
    const int mrow = mt * 16 + j + 8 * half;
    C[(size_t)mrow * N + nt * 16 + row] = c[j] + bs;
  }
}

// ---------------- fused triple-softmax + coda attention ----------------
// One block of 4 waves per (n, head, q-tile of 16).
__global__ __launch_bounds__(128) void attn_fused_kernel(
    const float* __restrict__ proj, const float* __restrict__ kten,
    const float* __restrict__ vten, const unsigned char* __restrict__ mask,
    float* __restrict__ mix) {
  __shared__ __align__(16) float sQ0[16 * HD];
  __shared__ __align__(16) float sQ1[16 * HD];
  __shared__ __align__(16) float sK[NWAVES][16 * HD];
  __shared__ __align__(16) float sAff[NWAVES][16 * 16];
  __shared__ unsigned sMax1[16];
  __shared__ float sSum1[16];
  __shared__ unsigned sMax2[16][NF];
  __shared__ float sSum2[16][NF];
  __shared__ unsigned sMax3[16][FJ + 3];
  __shared__ float sSum3[16][FJ + 3];

  const int tid = threadIdx.x;
  const int lane = tid & 31;
  const int wave = tid >> 5;
  const int half = lane >> 4;
  const int col = lane & 15;

  const int bid = blockIdx.x;
  const int qt = bid % QTILES;
  const int h = (bid / QTILES) % NH;
  const int n = bid / (QTILES * NH);

  // init stats
  const unsigned negord = f2ord(-3.0e38f);
  for (int i = tid; i < 16; i += 128) { sMax1[i] = negord; sSum1[i] = 0.f; }
  for (int i = tid; i < 16 * NF; i += 128) {
    (&sMax2[0][0])[i] = negord; (&sSum2[0][0])[i] = 0.f;
  }
  for (int i = tid; i < 16 * (FJ + 3); i += 128) {
    (&sMax3[0][0])[i] = negord; (&sSum3[0][0])[i] = 0.f;
  }

  // load q0/q1 tiles (proj row layout: [h*128 + act*64 + c]), zero-pad rows >= QL
  for (int i = tid; i < 16 * HD; i += 128) {
    const int r = i >> 6, c = i & 63;
    const int qr = qt * 16 + r;
    float a0 = 0.f, a1 = 0.f;
    if (qr < QL) {
      const float* row = proj + (size_t)(n * QL + qr) * PROJ_N + h * (2 * HD);
      a0 = row[c];
      a1 = row[HD + c];
    }
    sQ0[i] = a0;
    sQ1[i] = a1;
  }
  __syncthreads();

  const float* kbase = kten + ((size_t)n * KL * NH + h) * HD;
  const float* vbase = vten + ((size_t)n * KL * NH + h) * HD;
  const unsigned char* mrow = mask + (size_t)n * KL;

  // ===== pass 1: group maxes (full row / per-frame / per-in-frame-position) =====
  for (int it = 0; it < ITERS; ++it) {
    const int t = it * NWAVES + wave;
    const bool act = t < NKT;
    const int kk0 = t * 16;
    __syncthreads();
    if (act) stage_rows(kbase, kk0, sK[wave], lane);
    __syncthreads();
    if (act) {
      if (t + NWAVES < NKT)  // prefetch next tile for this wave into GL2
        __builtin_prefetch(kbase + (size_t)(kk0 + NWAVES * 16 + (lane & 15)) * (NH * HD), 0, 1);
      v8f s0 = qk_tile16(sQ0, sK[wave], lane);
      const int kk = kk0 + col;
      const bool mok = (kk < KL) && (mrow[kk < KL ? kk : 0] != 0);
      int f = 0, jj = 0;
      if (kk < KL) { f = kk / FJ; jj = kk - f * FJ; }
#pragma unroll
      for (int j = 0; j < 8; ++j) {
        const int q = j + 8 * half;
        const float sv = mok ? s0[j] * 0.125f : -3.0e38f;
        const unsigned o = f2ord(sv);
        atomicMax(&sMax1[q], o);
        atomicMax(&sMax2[q][f], o);
        atomicMax(&sMax3[q][jj], o);
      }
    }
  }
  __syncthreads();

  // ===== pass 2: exp-sums =====
  for (int it = 0; it < ITERS; ++it) {
    const int t = it * NWAVES + wave;
    const bool act = t < NKT;
    const int kk0 = t * 16;
    __syncthreads();
    if (act) stage_rows(kbase, kk0, sK[wave], lane);
    __syncthreads();
    if (act) {
      v8f s0 = qk_tile16(sQ0, sK[wave], lane);
      const int kk = kk0 + col;
      const bool mok = (kk < KL) && (mrow[kk < KL ? kk : 0] != 0);
      int f = 0, jj = 0;
      if (kk < KL) { f = kk / FJ; jj = kk - f * FJ; }
      if (mok) {
#pragma unroll
        for (int j = 0; j < 8; ++j) {
          const int q = j + 8 * half;
          const float sv = s0[j] * 0.125f;
          atomicAdd(&sSum1[q], __expf(sv - ord2f(sMax1[q])));
          atomicAdd(&sSum2[q][f], __expf(sv - ord2f(sMax2[q][f])));
          atomicAdd(&sSum3[q][jj], __expf(sv - ord2f(sMax3[q][jj])));
        }
      }
    }
  }
  __syncthreads();

  // ===== pass 3: normalize + coda gate + aff@v =====
  v8f macc[4];
#pragma unroll
  for (int ct = 0; ct < 4; ++ct) { v8f z = {}; macc[ct] = z; }

  for (int it = 0; it < ITERS; ++it) {
    const int t = it * NWAVES + wave;
    const bool act = t < NKT;
    const int kk0 = t * 16;
    __syncthreads();
    if (act) stage_rows(kbase, kk0, sK[wave], lane);
    __syncthreads();
    if (act) {
      if (t + NWAVES < NKT)
        __builtin_prefetch(vbase + (size_t)(kk0 + NWAVES * 16 + (lane & 15)) * (NH * HD), 0, 1);
      v8f s0 = qk_tile16(sQ0, sK[wave], lane);
      v8f s2 = qk_tile16(sQ1, sK[wave], lane);
      // L1 gate in C-matrix layout: lane -> (q = j + 8*half, k = col)
      float l1[8];
#pragma unroll
      for (int j = 0; j < 8; ++j) l1[j] = 0.f;
      const float4* krow = (const float4*)(sK[wave] + col * HD);
      for (int c4 = 0; c4 < 16; ++c4) {
        const float4 kv = krow[c4];
#pragma unroll
        for (int j = 0; j < 8; ++j) {
          const float4 qv = ((const float4*)(sQ1 + (j + 8 * half) * HD))[c4];
          l1[j] += fabsf(qv.x - kv.x) + fabsf(qv.y - kv.y) +
                   fabsf(qv.z - kv.z) + fabsf(qv.w - kv.w);
        }
      }
      const int kk = kk0 + col;
      const bool mok = (kk < KL) && (mrow[kk < KL ? kk : 0] != 0);
      int f = 0, jj = 0;
      if (kk < KL) { f = kk / FJ; jj = kk - f * FJ; }
#pragma unroll
      for (int j = 0; j < 8; ++j) {
        const int q = j + 8 * half;
        float affv = 0.f;
        if (mok) {
          const float sv = s0[j] * 0.125f;
          const float e1 = __expf(sv - ord2f(sMax1[q])) / sSum1[q];
          const float e2 = __expf(sv - ord2f(sMax2[q][f])) / sSum2[q][f];
          const float e3 = __expf(sv - ord2f(sMax3[q][jj])) / sSum3[q][jj];
          const float smx = (e1 + e2 + e3) * (1.0f / 3.0f);
          const float th = tanhf(s2[j] * 0.125f);
          const float g = 2.0f / (1.0f + __expf(l1[j] * 0.125f));
          affv = 0.5f * (smx + th * g);
        }
        sAff[wave][q * 16 + col] = affv;
      }
    }
    __syncthreads();
    if (act) stage_rows(vbase, kk0, sK[wave], lane);  // reuse k buffer for v
    __syncthreads();
    if (act) {
      const int off = half * 2;
      const float* av = sAff[wave] + col * 16;  // A row M = lane&15
      const float* vv = sK[wave];
#pragma unroll
      for (int ct = 0; ct < 4; ++ct) {
#pragma unroll
        for (int ks = 0; ks < 4; ++ks) {
          const int kloc = ks * 4 + off;
          v2f a = *(const v2f*)(av + kloc);
          v2f b;
          b.x = vv[kloc * HD + ct * 16 + col];
          b.y = vv[(kloc + 1) * HD + ct * 16 + col];
          macc[ct] = __builtin_amdgcn_wmma_f32_16x16x4_f32(
              false, a, false, b, (short)0, macc[ct], false, false);
        }
      }
    }
  }

  // ===== cross-wave reduction of partial mixes, store =====
  __syncthreads();
#pragma unroll
  for (int ct = 0; ct < 4; ++ct)
#pragma unroll
    for (int j = 0; j < 8; ++j)
      sK[wave][(j + 8 * half) * HD + ct * 16 + col] = macc[ct][j];
  __syncthreads();
  for (int i = tid; i < 16 * HD; i += 128) {
    const float s = sK[0][i] + sK[1][i] + sK[2][i] + sK[3][i];
    const int qr = qt * 16 + (i >> 6);
    if (qr < QL)
      mix[((size_t)(n * QL + qr)) * EMB + h * HD + (i & 63)] = s;
  }
}

extern "C" void kernel_launch(void* const* d_in, const int* in_sizes, int n_in,
                              void* d_out, int out_size, void* d_ws, size_t ws_size,
                              hipStream_t stream) {
  (void)in_sizes; (void)n_in; (void)out_size; (void)ws_size;
  const float* q = (const float*)d_in[0];
  const float* k = (const float*)d_in[1];
  const float* v = (const float*)d_in[2];
  const unsigned char* m = (const unsigned char*)d_in[3];
  const float* w_in = (const float*)d_in[4];
  const float* b_in = (const float*)d_in[5];
  const float* w_out = (const float*)d_in[6];
  const float* b_out = (const float*)d_in[7];
  float* out = (float*)d_out;

  float* proj = (float*)d_ws;                       // 1232 x 1536 f32
  float* mixb = proj + (size_t)NB * QL * PROJ_N;    // 1232 x 768  f32

  // proj = q @ w_in + b_in : 77 x 96 tiles, 8 waves/block
  gemm16_bias_kernel<<<(77 * 96) / 8, 256, 0, stream>>>(q, w_in, b_in, proj,
                                                        NB * QL, PROJ_N, EMB);
  // fused attention: one block per (n, head, q-tile)
  attn_fused_kernel<<<NB * NH * QTILES, 128, 0, stream>>>(proj, k, v, m, mixb);
  // out = mix @ w_out + b_out : 77 x 48 tiles
  gemm16_bias_kernel<<<(77 * 48) / 8, 256, 0, stream>>>(mixb, w_out, b_out, out,
                                                        NB * QL, EMB, EMB);
}